// JetGCN_48172353192134
// MI455X (gfx1250) — compile-verified
//
#include <hip/hip_runtime.h>
#include <hip/hip_bf16.h>

#define N_NODES  100000
#define N_EDGES  1200000
#define N_GRAPHS 512
#define HID      64

typedef __attribute__((ext_vector_type(2))) float v2f;
typedef __attribute__((ext_vector_type(8))) float v8f;

// ---------------------------------------------------------------- utilities
__global__ void set_one_kernel(float* __restrict__ p, int n) {
  int i = blockIdx.x * blockDim.x + threadIdx.x;
  if (i < n) p[i] = 1.0f;
}

__global__ void zero_kernel(float* __restrict__ p, int n) {
  int i = blockIdx.x * blockDim.x + threadIdx.x;
  if (i < n) p[i] = 0.0f;
}

// deg[d] += 1 per incoming edge (deg pre-initialized to 1.0 for self-loop)
__global__ void degree_kernel(const int* __restrict__ dst, float* __restrict__ deg, int ne) {
  int e = blockIdx.x * blockDim.x + threadIdx.x;
  if (e < ne) atomicAdd(&deg[dst[e]], 1.0f);
}

// deg -> deg^{-1/2} in place (deg >= 1 always due to self-loops)
__global__ void rsqrt_kernel(float* __restrict__ p, int n) {
  int i = blockIdx.x * blockDim.x + threadIdx.x;
  if (i < n) p[i] = rsqrtf(p[i]);
}

// ------------------------------------------------------------ WMMA GEMM
// C[nrows x 64] = A[nrows x K] @ W[K x 64]; K % 4 == 0, nrows % 16 == 0.
// One wave per 16x16 output tile; 4 waves per block cover n = 0..63.
// f32 WMMA operand layouts per CDNA5 ISA 7.12.2:
//   A 16x4 : VGPR0 = K=0 (lanes 0-15) / K=2 (lanes 16-31); VGPR1 = K=1 / K=3
//   B 4x16 : VGPR0 = row K=0 (lanes 0-15) / K=2 (lanes 16-31); VGPR1 = K=1 / K=3
//   C/D    : VGPR j = row (j + 8*half), N = lane & 15
__global__ __launch_bounds__(128) void gemm_wmma_kernel(
    const float* __restrict__ A, const float* __restrict__ W,
    float* __restrict__ C, int K) {
  const int lane = threadIdx.x & 31;
  const int wave = threadIdx.x >> 5;
  const int m0   = blockIdx.x << 4;
  const int n0   = wave << 4;
  const int half = lane >> 4;
  const int l15  = lane & 15;
  const int m    = m0 + l15;

  v8f acc = {};
  for (int k0 = 0; k0 < K; k0 += 4) {
    const int ka = k0 + (half << 1);
    v2f a, b;
    a.x = A[m * K + ka];
    a.y = A[m * K + ka + 1];
    b.x = W[ka * HID + n0 + l15];
    b.y = W[(ka + 1) * HID + n0 + l15];
    acc = __builtin_amdgcn_wmma_f32_16x16x4_f32(
        /*neg_a=*/false, a, /*neg_b=*/false, b,
        /*c_mod=*/(short)0, acc, /*reuse_a=*/false, /*reuse_b=*/false);
  }

  const int c0 = (m0 + (half << 3)) * HID + n0 + l15;
#pragma unroll
  for (int j = 0; j < 8; ++j) C[c0 + j * HID] = acc[j];
}

// ------------------------------------------------------- edge aggregation
// One lane per (edge, feature); lanes of a wave = consecutive features of
// one edge -> coalesced h-load and coalesced atomic burst into L2.
__global__ void scatter_kernel(const float* __restrict__ h,
                               const int* __restrict__ src,
                               const int* __restrict__ dst,
                               const float* __restrict__ dinv,
                               float* __restrict__ agg) {
  long long tid = (long long)blockIdx.x * blockDim.x + threadIdx.x;
  int e = (int)(tid >> 6);
  int f = (int)(tid & 63);
  if (e >= N_EDGES) return;
  int s = src[e];
  int d = dst[e];
  float nrm = dinv[s] * dinv[d];
  atomicAdd(&agg[d * HID + f], h[s * HID + f] * nrm);
}

// out = relu(agg + h_self * dinv^2 + b), written into agg in place.
// Folds the self-loop term (norm = dinv[i]*dinv[i]) without atomics.
__global__ void bias_relu_kernel(float* __restrict__ agg,
                                 const float* __restrict__ hself,
                                 const float* __restrict__ dinv,
                                 const float* __restrict__ b) {
  long long tid = (long long)blockIdx.x * blockDim.x + threadIdx.x;
  int i = (int)(tid >> 6);
  int f = (int)(tid & 63);
  if (i >= N_NODES) return;
  float di = dinv[i];
  float v = agg[i * HID + f] + hself[i * HID + f] * (di * di) + b[f];
  agg[i * HID + f] = fmaxf(v, 0.0f);
}

// ------------------------------------------------------------ mean pool
__global__ void pool_kernel(const float* __restrict__ h,
                            const int* __restrict__ batch,
                            float* __restrict__ gsum,
                            float* __restrict__ gcnt) {
  long long tid = (long long)blockIdx.x * blockDim.x + threadIdx.x;
  int i = (int)(tid >> 6);
  int f = (int)(tid & 63);
  if (i >= N_NODES) return;
  int g = batch[i];
  atomicAdd(&gsum[g * HID + f], h[i * HID + f]);
  if (f == 0) atomicAdd(&gcnt[g], 1.0f);
}

// out[g][o] = (sum_k gsum[g][k] * Wl[k][o]) / max(cnt,1) + bl[o]
__global__ void final_kernel(const float* __restrict__ gsum,
                             const float* __restrict__ gcnt,
                             const float* __restrict__ Wl,
                             const float* __restrict__ bl,
                             float* __restrict__ out) {
  int tid = blockIdx.x * blockDim.x + threadIdx.x;
  if (tid >= N_GRAPHS * 2) return;
  int g = tid >> 1;
  int o = tid & 1;
  float c = fmaxf(gcnt[g], 1.0f);
  float s = 0.0f;
#pragma unroll 8
  for (int k = 0; k < HID; ++k) s += gsum[g * HID + k] * Wl[k * 2 + o];
  out[tid] = s / c + bl[o];
}

// ------------------------------------------------------------------ launch
extern "C" void kernel_launch(void* const* d_in, const int* in_sizes, int n_in,
                              void* d_out, int out_size, void* d_ws, size_t ws_size,
                              hipStream_t stream) {
  (void)in_sizes; (void)n_in; (void)out_size; (void)ws_size;

  const float* x   = (const float*)d_in[0];
  const int*   ei  = (const int*)d_in[1];
  const int*   bat = (const int*)d_in[2];
  const float* W1  = (const float*)d_in[3];
  const float* b1  = (const float*)d_in[4];
  const float* W2  = (const float*)d_in[5];
  const float* b2  = (const float*)d_in[6];
  const float* Wl  = (const float*)d_in[7];
  const float* bl  = (const float*)d_in[8];
  float* out = (float*)d_out;

  const int* src = ei;            // edge_index[0]
  const int* dst = ei + N_EDGES;  // edge_index[1]

  // workspace layout (~52 MB)
  float* deg  = (float*)d_ws;                       // N_NODES (becomes dinv)
  float* bufA = deg  + 100096;                      // N_NODES*64
  float* bufB = bufA + (size_t)N_NODES * HID;       // N_NODES*64
  float* gsum = bufB + (size_t)N_NODES * HID;       // 512*64
  float* gcnt = gsum + N_GRAPHS * HID;              // 512  (contiguous after gsum)

  auto cdiv = [](long long a, long long b) { return (unsigned)((a + b - 1) / b); };
  const long long NF = (long long)N_NODES * HID;
  const long long EF = (long long)N_EDGES * HID;

  // degrees + normalization
  set_one_kernel<<<cdiv(N_NODES, 256), 256, 0, stream>>>(deg, N_NODES);
  zero_kernel<<<cdiv(NF, 256), 256, 0, stream>>>(bufB, (int)NF);
  zero_kernel<<<cdiv(N_GRAPHS * HID + N_GRAPHS, 256), 256, 0, stream>>>(
      gsum, N_GRAPHS * HID + N_GRAPHS);
  degree_kernel<<<cdiv(N_EDGES, 256), 256, 0, stream>>>(dst, deg, N_EDGES);
  rsqrt_kernel<<<cdiv(N_NODES, 256), 256, 0, stream>>>(deg, N_NODES);

  // layer 1: h1 = x @ W1 ; aggregate ; relu(+bias+self-loop)  -> bufB
  gemm_wmma_kernel<<<N_NODES / 16, 128, 0, stream>>>(x, W1, bufA, 4);
  scatter_kernel<<<cdiv(EF, 256), 256, 0, stream>>>(bufA, src, dst, deg, bufB);
  bias_relu_kernel<<<cdiv(NF, 256), 256, 0, stream>>>(bufB, bufA, deg, b1);

  // layer 2: h2 = h1relu @ W2 ; aggregate ; relu  -> bufB
  gemm_wmma_kernel<<<N_NODES / 16, 128, 0, stream>>>(bufB, W2, bufA, HID);
  zero_kernel<<<cdiv(NF, 256), 256, 0, stream>>>(bufB, (int)NF);
  scatter_kernel<<<cdiv(EF, 256), 256, 0, stream>>>(bufA, src, dst, deg, bufB);
  bias_relu_kernel<<<cdiv(NF, 256), 256, 0, stream>>>(bufB, bufA, deg, b2);

  // mean pool + linear head
  pool_kernel<<<cdiv(NF, 256), 256, 0, stream>>>(bufB, bat, gsum, gcnt);
  final_kernel<<<cdiv(N_GRAPHS * 2, 256), 256, 0, stream>>>(gsum, gcnt, Wl, bl, out);
}